// VectorQuantiser_67525475827789
// MI455X (gfx1250) — compile-verified
//
#include <hip/hip_runtime.h>

typedef __attribute__((ext_vector_type(2))) float v2f;
typedef __attribute__((ext_vector_type(8))) float v8f;
typedef __attribute__((ext_vector_type(4))) int   v4i;

#define N_E      1024
#define E_DIM    256
#define BETA     0.2f
#define SPATIAL  32768          // 32*32*32
#define TOKENS   65536          // 2 * SPATIAL
#define ZTOT     16777216       // 2 * 256 * SPATIAL

// d_out layout (floats, concatenated in return order)
#define OFF_LOSS 16777216
#define OFF_IDX  16777217
#define OFF_SAMP 16842753       // OFF_IDX + 65536
#define OFF_MCD  16844801       // OFF_SAMP + 2048
#define OFF_MCV  16844802

// workspace layout (floats)
#define WS_SSE   0              // 32768 block partials
#define WS_NRM   32768          // 1024 ||emb||^2
#define WS_CM2   33792          // 1024 per-col 2nd-min
#define WS_CS1   34816          // 1024 per-col sum
#define WS_CS2   35840          // 1024 per-col sum-of-squares

#if __has_builtin(__builtin_amdgcn_global_load_async_to_lds_b128) && \
    __has_builtin(__builtin_amdgcn_s_wait_asynccnt)
#define ASYNC_LDS 1
typedef __attribute__((address_space(1))) v4i* g_v4i_ptr;
typedef __attribute__((address_space(3))) v4i* l_v4i_ptr;
#else
#define ASYNC_LDS 0
#endif

// ---------------------------------------------------------------- emb norms
__global__ __launch_bounds__(256) void k_emb_norms(
    const float* __restrict__ emb, float* __restrict__ norms) {
  const int row  = blockIdx.x * 8 + (threadIdx.x >> 5);
  const int lane = threadIdx.x & 31;
  const float* e = emb + (size_t)row * E_DIM;
  float s = 0.f;
  for (int c = lane; c < E_DIM; c += 32) s += e[c] * e[c];
#pragma unroll
  for (int off = 16; off > 0; off >>= 1) s += __shfl_xor(s, off, 32);
  if (lane == 0) norms[row] = s;
}

// ----------------------------------------------------- LDS chunk staging
// One chunk = 16 codes x 256 f32 = 16 KB.
__device__ __forceinline__ void stage_chunk(
    const float* __restrict__ emb, const float* __restrict__ norms,
    float* ldsbuf, float* ldsnrm, int chunk) {
#if ASYNC_LDS
  float* src = const_cast<float*>(emb) + (size_t)chunk * 16 * E_DIM;
  int i = threadIdx.x;                     // 1024 x b128, 4 per thread
#pragma unroll
  for (int r = 0; r < 4; ++r, i += 256) {
    __builtin_amdgcn_global_load_async_to_lds_b128(
        (g_v4i_ptr)(src + i * 4),
        (l_v4i_ptr)(ldsbuf + i * 4),
        0, 0);
  }
#else
  const float4* s4 = (const float4*)(emb + (size_t)chunk * 16 * E_DIM);
  float4* d4 = (float4*)ldsbuf;
  for (int i = threadIdx.x; i < (16 * E_DIM) / 4; i += 256) d4[i] = s4[i];
  if (chunk + 1 < 64)
    __builtin_prefetch(emb + (size_t)(chunk + 1) * 16 * E_DIM, 0, 1);
#endif
  if (threadIdx.x < 16) ldsnrm[threadIdx.x] = norms[chunk * 16 + threadIdx.x];
}

// ------------------------------------------------- distance + argmin (WMMA)
// D = emb(M=16 codes) x z^T(N=16 tokens), K=256 via v_wmma_f32_16x16x4_f32.
// Per wave: 16 tokens vs all 1024 codes; emb double-buffered in LDS via
// async global->LDS loads; two independent K-split accumulator chains.
__global__ __launch_bounds__(256) void k_dist_argmin(
    const float* __restrict__ z, const float* __restrict__ emb,
    const float* __restrict__ norms, float* __restrict__ idx_out) {
  __shared__ __align__(16) float lds_emb[2][16 * E_DIM];  // 2 x 16 KB
  __shared__ float lds_nrm[2][16];

  const int lane = threadIdx.x & 31;
  const int wave = threadIdx.x >> 5;
  const int t0   = (blockIdx.x * 8 + wave) * 16;
  const int m    = lane & 15;
  const int koff = (lane < 16) ? 0 : 2;

  // B fragment source: token t0+m, features strided by SPATIAL (channel-major z)
  const int tok = t0 + m;
  const int b   = tok >> 15;
  const int s   = tok & (SPATIAL - 1);
  const float* zb = z + (size_t)b * E_DIM * SPATIAL + s;

  v2f bfrag[64];
#pragma unroll
  for (int k = 0; k < 64; ++k) {
    const int c = 4 * k + koff;
    v2f t;
    t.x = zb[(size_t)c * SPATIAL];
    t.y = zb[(size_t)(c + 1) * SPATIAL];
    bfrag[k] = t;
  }

  float bestD = 3.4e38f;
  int   bestI = 0;

  stage_chunk(emb, norms, lds_emb[0], lds_nrm[0], 0);

  for (int cb = 0; cb < 64; ++cb) {
    const int cur = cb & 1;
    if (cb + 1 < 64) {
      stage_chunk(emb, norms, lds_emb[cur ^ 1], lds_nrm[cur ^ 1], cb + 1);
#if ASYNC_LDS
      __builtin_amdgcn_s_wait_asynccnt(4);   // current chunk's 4 loads done
#endif
    } else {
#if ASYNC_LDS
      __builtin_amdgcn_s_wait_asynccnt(0);
#endif
    }
    __syncthreads();                         // chunk cb visible to all waves

    const float* cbuf = lds_emb[cur];
    v8f acc0 = {0.f, 0.f, 0.f, 0.f, 0.f, 0.f, 0.f, 0.f};
    v8f acc1 = {0.f, 0.f, 0.f, 0.f, 0.f, 0.f, 0.f, 0.f};
#pragma unroll
    for (int k = 0; k < 32; ++k) {
      v2f a0 = *(const v2f*)&cbuf[m * E_DIM + 4 * k + koff];
      v2f a1 = *(const v2f*)&cbuf[m * E_DIM + 4 * (k + 32) + koff];
      acc0 = __builtin_amdgcn_wmma_f32_16x16x4_f32(
          false, a0, false, bfrag[k], (short)0, acc0, false, false);
      acc1 = __builtin_amdgcn_wmma_f32_16x16x4_f32(
          false, a1, false, bfrag[k + 32], (short)0, acc1, false, false);
    }
    const int mBase = (lane < 16) ? 0 : 8;
#pragma unroll
    for (int v = 0; v < 8; ++v) {
      const float d = lds_nrm[cur][mBase + v] - 2.0f * (acc0[v] + acc1[v]);
      const int code = cb * 16 + mBase + v;
      if (d < bestD) { bestD = d; bestI = code; }
    }
    __syncthreads();                         // all waves done with buf[cur]
  }

  // lanes n and n+16 hold the same token with disjoint code subsets: merge
  const float oD = __shfl_xor(bestD, 16, 32);
  const int   oI = __shfl_xor(bestI, 16, 32);
  if (oD < bestD || (oD == bestD && oI < bestI)) { bestD = oD; bestI = oI; }
  if (lane < 16) idx_out[t0 + m] = (float)bestI;
}

// ------------------------------------------ gather + z_q store + SSE partial
__global__ __launch_bounds__(512) void k_gather_sse(
    const float* __restrict__ z, const float* __restrict__ emb,
    const float* __restrict__ idxf, float* __restrict__ zq,
    float* __restrict__ sse_part) {
  __shared__ float sm[512];
  const size_t gid = (size_t)blockIdx.x * 512 + threadIdx.x;
  const int c = (int)((gid >> 15) & 255);
  const int b = (int)(gid >> 23);
  const int s = (int)(gid & (SPATIAL - 1));
  const int t = (b << 15) | s;
  const int i = (int)idxf[t];
  const float v = emb[(size_t)i * E_DIM + c];
  const float d = v - z[gid];
  zq[gid] = v;                       // straight-through: z_q_out == emb[idx]
  sm[threadIdx.x] = d * d;
  __syncthreads();
  for (int st = 256; st > 0; st >>= 1) {
    if (threadIdx.x < st) sm[threadIdx.x] += sm[threadIdx.x + st];
    __syncthreads();
  }
  if (threadIdx.x == 0) sse_part[blockIdx.x] = sm[0];
}

// ----------------------------------------------------------- sampled index
__global__ void k_zero_sampled(float* __restrict__ samp) {
  samp[blockIdx.x * 256 + threadIdx.x] = 0.f;
}
__global__ void k_sampled(const float* __restrict__ idxf, float* __restrict__ samp) {
  const int t = blockIdx.x * 256 + threadIdx.x;
  samp[(int)idxf[t]] = 1.0f;  // benign race: all writers store 1.0
}

// ------------------------------------------- codebook self-distances (WMMA)
// Each block owns 16 columns j; 8 waves sweep the 64 row tiles. cd symmetric,
// so per-column sum/sum^2 double as the reference's per-row stats.
__global__ __launch_bounds__(256) void k_cb_stats(
    const float* __restrict__ emb, const float* __restrict__ norms,
    float* __restrict__ col_min2, float* __restrict__ col_s1,
    float* __restrict__ col_s2) {
  __shared__ float rm1[8][32], rm2[8][32], rs1[8][32], rs2[8][32];
  const int lane  = threadIdx.x & 31;
  const int wave  = threadIdx.x >> 5;
  const int jBase = blockIdx.x * 16;
  const int m     = lane & 15;
  const int koff  = (lane < 16) ? 0 : 2;
  const int j     = jBase + m;
  const float nj  = norms[j];

  v2f bfrag[64];
#pragma unroll
  for (int k = 0; k < 64; ++k)
    bfrag[k] = *(const v2f*)&emb[(size_t)j * E_DIM + 4 * k + koff];

  float m1 = 3.4e38f, m2 = 3.4e38f, s1 = 0.f, s2 = 0.f;

  for (int it = wave; it < 64; it += 8) {
    const int rowBase = it * 16;
    const int i0 = rowBase + m;
    v8f acc = {0.f, 0.f, 0.f, 0.f, 0.f, 0.f, 0.f, 0.f};
#pragma unroll
    for (int k = 0; k < 64; ++k) {
      v2f afrag = *(const v2f*)&emb[(size_t)i0 * E_DIM + 4 * k + koff];
      acc = __builtin_amdgcn_wmma_f32_16x16x4_f32(
          false, afrag, false, bfrag[k], (short)0, acc, false, false);
    }
    const int rOff = rowBase + ((lane < 16) ? 0 : 8);
#pragma unroll
    for (int v = 0; v < 8; ++v) {
      const float val = norms[rOff + v] + nj - 2.0f * acc[v];
      if (val < m1)      { m2 = m1; m1 = val; }
      else if (val < m2) { m2 = val; }
      s1 += val;
      s2 += val * val;
    }
  }
  rm1[wave][lane] = m1; rm2[wave][lane] = m2;
  rs1[wave][lane] = s1; rs2[wave][lane] = s2;
  __syncthreads();
  if (threadIdx.x < 32) {
    float a1 = rm1[0][lane], a2 = rm2[0][lane];
    float t1 = rs1[0][lane], t2 = rs2[0][lane];
    for (int w = 1; w < 8; ++w) {
      const float b1 = rm1[w][lane], b2 = rm2[w][lane];
      const float n1 = fminf(a1, b1);
      const float n2 = fminf(fmaxf(a1, b1), fminf(a2, b2));
      a1 = n1; a2 = n2;
      t1 += rs1[w][lane]; t2 += rs2[w][lane];
    }
    const float o1 = __shfl_xor(a1, 16, 32), o2 = __shfl_xor(a2, 16, 32);
    const float u1 = __shfl_xor(t1, 16, 32), u2 = __shfl_xor(t2, 16, 32);
    const float f2 = fminf(fmaxf(a1, o1), fminf(a2, o2));
    t1 += u1; t2 += u2;
    if (lane < 16) {
      col_min2[jBase + lane] = f2;
      col_s1[jBase + lane]   = t1;
      col_s2[jBase + lane]   = t2;
    }
  }
}

// ----------------------------------------------------------------- finalize
__global__ __launch_bounds__(256) void k_finalize(
    const float* __restrict__ sse_part, const float* __restrict__ col_min2,
    const float* __restrict__ col_s1, const float* __restrict__ col_s2,
    float* __restrict__ out) {
  __shared__ float sm[256];
  const int tid = threadIdx.x;

  float a = 0.f;
  for (int i = tid; i < 32768; i += 256) a += sse_part[i];
  sm[tid] = a; __syncthreads();
  for (int st = 128; st > 0; st >>= 1) { if (tid < st) sm[tid] += sm[tid + st]; __syncthreads(); }
  const float sse = sm[0];
  __syncthreads();

  float md = 0.f;
  for (int i = tid; i < N_E; i += 256) md += col_min2[i];
  sm[tid] = md; __syncthreads();
  for (int st = 128; st > 0; st >>= 1) { if (tid < st) sm[tid] += sm[tid + st]; __syncthreads(); }
  const float mcd = sm[0] / (float)N_E;
  __syncthreads();

  float mv = 0.f;
  for (int i = tid; i < N_E; i += 256) {
    const float s1 = col_s1[i], s2 = col_s2[i];
    mv += (s2 - s1 * s1 / (float)N_E) / (float)(N_E - 1);
  }
  sm[tid] = mv; __syncthreads();
  for (int st = 128; st > 0; st >>= 1) { if (tid < st) sm[tid] += sm[tid + st]; __syncthreads(); }
  const float mcv = sm[0] / (float)N_E;

  if (tid == 0) {
    out[OFF_LOSS] = (1.0f + BETA) * sse / (float)ZTOT;
    out[OFF_MCD]  = mcd;
    out[OFF_MCV]  = mcv;
  }
}

extern "C" void kernel_launch(void* const* d_in, const int* in_sizes, int n_in,
                              void* d_out, int out_size, void* d_ws, size_t ws_size,
                              hipStream_t stream) {
  const float* z   = (const float*)d_in[0];
  const float* emb = (const float*)d_in[1];
  float* out = (float*)d_out;
  float* ws  = (float*)d_ws;

  float* sse_part = ws + WS_SSE;
  float* norms    = ws + WS_NRM;
  float* cm2      = ws + WS_CM2;
  float* cs1      = ws + WS_CS1;
  float* cs2      = ws + WS_CS2;

  float* idxf = out + OFF_IDX;   // argmin indices stored as float (exact < 2^24)
  float* samp = out + OFF_SAMP;

  k_emb_norms  <<<128,   256, 0, stream>>>(emb, norms);
  k_dist_argmin<<<512,   256, 0, stream>>>(z, emb, norms, idxf);
  k_gather_sse <<<32768, 512, 0, stream>>>(z, emb, idxf, out, sse_part);
  k_zero_sampled<<<8,    256, 0, stream>>>(samp);
  k_sampled    <<<256,   256, 0, stream>>>(idxf, samp);
  k_cb_stats   <<<64,    256, 0, stream>>>(emb, norms, cm2, cs1, cs2);
  k_finalize   <<<1,     256, 0, stream>>>(sse_part, cm2, cs1, cs2, out);
}